// VectorQuantizer_12902081757497
// MI455X (gfx1250) — compile-verified
//
#include <hip/hip_runtime.h>

typedef __attribute__((ext_vector_type(2))) float v2f;
typedef __attribute__((ext_vector_type(4))) float v4f;
typedef __attribute__((ext_vector_type(8))) float v8f;

#define K_CODES   1024
#define DIM       64
#define HW        1024
#define NB        128           // n rows per block
#define KCHUNK    128           // codebook rows staged in LDS per chunk
#define LDS_STRIDE 68           // padded row stride (floats): conflict-free b64 reads, 16B-aligned rows

#define ZQ_OFF    0
#define LOSS_OFF  4194304       // B*D*H*W
#define IDX_OFF   4194305
#define USAGE_OFF 4259841       // IDX_OFF + 65536
#define N_ELEMS   4194304.0f

// ws float layout: [0..1023] e_norms, [1024..2047] used flags, [2048] loss accumulator

__global__ __launch_bounds__(256) void vq_pre(const float* __restrict__ emb,
                                              float* __restrict__ ws) {
  int gid = blockIdx.x * 256 + threadIdx.x;
  if (gid < K_CODES) {
    const v4f* row = (const v4f*)(emb + gid * DIM);
    float s = 0.f;
#pragma unroll
    for (int i = 0; i < DIM / 4; ++i) {
      v4f v = row[i];
      s += v.x * v.x + v.y * v.y + v.z * v.z + v.w * v.w;
    }
    ws[gid] = s;
    ws[K_CODES + gid] = 0.f;
    if (gid == 0) ws[2 * K_CODES] = 0.f;
  }
}

__global__ __launch_bounds__(256) void vq_main(const float* __restrict__ ze,
                                               const float* __restrict__ emb,
                                               float* __restrict__ out,
                                               float* __restrict__ ws) {
  __shared__ float xs[NB * LDS_STRIDE];      // x tile, [n_local][d]
  __shared__ float es[KCHUNK * LDS_STRIDE];  // codebook chunk, [k_local][d]
  __shared__ int   idx_s[NB];

  const int tid  = threadIdx.x;
  const int lane = tid & 31;
  const int wave = tid >> 5;
  const int col  = lane & 15;          // N index within 16-wide tile
  const int dof  = (lane >> 4) << 1;   // K sub-offset: 0 for lanes 0-15, 2 for 16-31

  const int n0  = blockIdx.x * NB;     // block's first flat n (= b*1024 + hw)
  const int b   = n0 >> 10;
  const int hw0 = n0 & (HW - 1);

  // Stage z_e tile (128 n x 64 d) into LDS as xs[n_local][d]; global reads coalesced b128.
  for (int it = 0; it < 8; ++it) {
    int v  = it * 256 + tid;           // 0..2047 float4 slots
    int d  = v >> 5;                   // 32 float4 per d-slice
    int nl = (v & 31) << 2;
    v4f val = *(const v4f*)(ze + (((size_t)(b * DIM + d)) << 10) + hw0 + nl);
    xs[(nl + 0) * LDS_STRIDE + d] = val.x;
    xs[(nl + 1) * LDS_STRIDE + d] = val.y;
    xs[(nl + 2) * LDS_STRIDE + d] = val.z;
    xs[(nl + 3) * LDS_STRIDE + d] = val.w;
  }
  __syncthreads();

  // A fragments: wave's 16 rows, 16 steps of K=4 (D=64)
  v2f a[16];
  {
    const float* xrow = &xs[(wave * 16 + col) * LDS_STRIDE + dof];
#pragma unroll
    for (int s = 0; s < 16; ++s) a[s] = *(const v2f*)(xrow + (s << 2));
  }

  float minv[8];
  int   mini[8];
#pragma unroll
  for (int r = 0; r < 8; ++r) { minv[r] = 3.4e38f; mini[r] = 0; }

  for (int k0 = 0; k0 < K_CODES; k0 += KCHUNK) {
    __syncthreads();   // previous chunk fully consumed before overwrite
    // Cooperative stage of 128 codebook rows; rows are contiguous in global,
    // LDS row starts are 16B-aligned (68*4B = 272 = 17*16) -> single b128 stores.
    for (int it = 0; it < 8; ++it) {
      int v  = it * 256 + tid;         // 0..2047 float4 slots
      int kl = v >> 4;                 // 16 float4 per code row
      int d4 = (v & 15) << 2;
      v4f val = *(const v4f*)(emb + ((size_t)(k0 + kl) << 6) + d4);
      *(v4f*)(&es[kl * LDS_STRIDE + d4]) = val;
    }
    __syncthreads();

    // hoist ||e||^2 for all 8 tiles of this chunk (no loadcnt waits inside wmma loop)
    float en[8];
#pragma unroll
    for (int j = 0; j < 8; ++j) en[j] = ws[k0 + j * 16 + col];

#pragma unroll
    for (int j = 0; j < 8; ++j) {
      const float* eb = &es[(j * 16 + col) * LDS_STRIDE + dof];
      v2f bf[16];
#pragma unroll
      for (int s = 0; s < 16; ++s) bf[s] = *(const v2f*)(eb + (s << 2));

      v8f c = {0.f, 0.f, 0.f, 0.f, 0.f, 0.f, 0.f, 0.f};
#pragma unroll
      for (int s = 0; s < 16; ++s)
        c = __builtin_amdgcn_wmma_f32_16x16x4_f32(false, a[s], false, bf[s],
                                                  (short)0, c, false, false);
      // dist-equivalent: ||e||^2 - 2 x.e  (||x||^2 constant per row)
#pragma unroll
      for (int r = 0; r < 8; ++r) {
        float dv = fmaf(-2.0f, c[r], en[j]);
        if (dv < minv[r]) { minv[r] = dv; mini[r] = k0 + j * 16 + col; }
      }
    }
  }

  // lexicographic (val, idx) min-reduce across each 16-lane half
#pragma unroll
  for (int r = 0; r < 8; ++r) {
    float v  = minv[r];
    int   ki = mini[r];
#pragma unroll
    for (int m = 1; m <= 8; m <<= 1) {
      float ov = __shfl_xor(v, m, 32);
      int   ok = __shfl_xor(ki, m, 32);
      if (ov < v || (ov == v && ok < ki)) { v = ov; ki = ok; }
    }
    // lanes<16 hold row r of tile, lanes>=16 hold row 8+r
    if (col == 0) idx_s[wave * 16 + (lane >> 4) * 8 + r] = ki;
  }
  __syncthreads();

  // indices output (as float) + usage marks (idempotent stores)
  if (tid < NB) {
    int k = idx_s[tid];
    out[IDX_OFF + n0 + tid] = (float)k;
    ws[K_CODES + k] = 1.0f;
  }

  // z_q scatter-back (coalesced per d-slice) + commitment-loss partial from LDS x tile
  float lsum = 0.f;
  for (int it = 0; it < 32; ++it) {
    int v  = it * 256 + tid;           // 0..8191
    int d  = v >> 7;
    int nl = v & 127;
    int k  = idx_s[nl];
    float ev = emb[((size_t)k << 6) + d];
    float xv = xs[nl * LDS_STRIDE + d];
    float df = xv - ev;
    lsum += df * df;
    out[ZQ_OFF + (((size_t)(b * DIM + d)) << 10) + hw0 + nl] = ev;
  }
#pragma unroll
  for (int m = 1; m < 32; m <<= 1) lsum += __shfl_xor(lsum, m, 32);
  if (lane == 0) atomicAdd(&ws[2 * K_CODES], lsum);
}

__global__ __launch_bounds__(256) void vq_fin(const float* __restrict__ ws,
                                              float* __restrict__ out) {
  __shared__ float red[8];
  int tid = threadIdx.x;
  float s = 0.f;
  for (int i = tid; i < K_CODES; i += 256) s += ws[K_CODES + i];
#pragma unroll
  for (int m = 1; m < 32; m <<= 1) s += __shfl_xor(s, m, 32);
  if ((tid & 31) == 0) red[tid >> 5] = s;
  __syncthreads();
  if (tid == 0) {
    float t = 0.f;
#pragma unroll
    for (int i = 0; i < 8; ++i) t += red[i];
    out[USAGE_OFF] = t / (float)K_CODES;
    out[LOSS_OFF]  = ws[2 * K_CODES] / N_ELEMS;
  }
}

extern "C" void kernel_launch(void* const* d_in, const int* in_sizes, int n_in,
                              void* d_out, int out_size, void* d_ws, size_t ws_size,
                              hipStream_t stream) {
  const float* z_e = (const float*)d_in[0];
  const float* emb = (const float*)d_in[1];
  float* out = (float*)d_out;
  float* ws  = (float*)d_ws;

  vq_pre<<<4, 256, 0, stream>>>(emb, ws);
  vq_main<<<65536 / NB, 256, 0, stream>>>(z_e, emb, out, ws);
  vq_fin<<<1, 256, 0, stream>>>(ws, out);
}